// LSTM_58918361366817
// MI455X (gfx1250) — compile-verified
//
#include <hip/hip_runtime.h>
#include <math.h>

#define T_STEPS 128
#define HDIM    512
#define NB_SEQ  32      // persistent blocks: 32 x 16 = 512 h-slices

typedef __attribute__((ext_vector_type(2))) float v2f;
typedef __attribute__((ext_vector_type(8))) float v8f;

// Partial matvec of 16 consecutive rows (Wrows, row stride HDIM) against vector v,
// over k in [k0, k0+klen), using V_WMMA_F32_16X16X4_F32 with B = v broadcast into
// all 16 columns. Returns accumulator in WMMA D layout.
__device__ __forceinline__ v8f wmma_matvec(const float* __restrict__ Wrows,
                                           const float* __restrict__ v,
                                           int k0, int klen)
{
    const int lane = threadIdx.x & 31;          // wave32
    const int m    = lane & 15;                 // A row within tile
    const int kb   = (lane >> 4) << 1;          // 0 (lanes 0-15) or 2 (lanes 16-31)
    const float* wp = Wrows + (size_t)m * HDIM + k0 + kb;
    const float* vp = v + k0 + kb;
    v8f acc0 = {};
    v8f acc1 = {};
    #pragma unroll 4
    for (int k = 0; k < klen; k += 8) {
        v2f a0 = *(const v2f*)(wp + k);
        v2f b0 = *(const v2f*)(vp + k);
        acc0 = __builtin_amdgcn_wmma_f32_16x16x4_f32(false, a0, false, b0,
                                                     (short)0, acc0, false, false);
        v2f a1 = *(const v2f*)(wp + k + 4);
        v2f b1 = *(const v2f*)(vp + k + 4);
        acc1 = __builtin_amdgcn_wmma_f32_16x16x4_f32(false, a1, false, b1,
                                                     (short)0, acc1, false, false);
    }
    return acc0 + acc1;
}

// D layout: vgpr j, lanes 0-15 -> M=j (N=lane), lanes 16-31 -> M=j+8.
// Column N==0 (lanes 0 and 16) carries the full 16-row result.
__device__ __forceinline__ void store_col0(float* dst16, v8f acc)
{
    const int lane = threadIdx.x & 31;
    if ((lane & 15) == 0) {
        const int mb = (lane >> 4) * 8;
        #pragma unroll
        for (int i = 0; i < 8; ++i) dst16[mb + i] = acc[i];
    }
}

// ---------------- Kernel 1: Zx[t][g][j] = Wx[t,g,j,:].x[t,:] + b[t,g,j] ----------------
// grid: T*32 blocks, 256 threads (8 waves). Block = (t, 16-row h-slice j0).
// Wave w: gate = w&3, K-half = w>>2 (256 each). Streams 512 MB fully in parallel.
__global__ void zx_kernel(const float* __restrict__ x,
                          const float* __restrict__ Wx,
                          const float* __restrict__ b,
                          float* __restrict__ Zx)
{
    __shared__ float part[8][16];
    const int t  = blockIdx.x >> 5;
    const int j0 = (blockIdx.x & 31) << 4;
    const int w  = threadIdx.x >> 5;
    const int g  = w & 3;
    const int kh = w >> 2;

    const float* Wrows = Wx + (((size_t)t * 4 + g) * HDIM + j0) * HDIM;
    const float* v     = x + (size_t)t * HDIM;

    v8f acc = wmma_matvec(Wrows, v, kh * 256, 256);
    store_col0(part[w], acc);
    __syncthreads();

    if (threadIdx.x < 64) {
        const int g2 = threadIdx.x >> 4;
        const int m  = threadIdx.x & 15;
        const size_t o = ((size_t)t * 4 + g2) * HDIM + j0 + m;
        Zx[o] = part[g2][m] + part[g2 + 4][m] + b[o];
    }
}

// ---------------- Kernel 2: persistent sequential LSTM + final projection ----------------
// 32 blocks x 256 threads. Block owns h/c slice [j0, j0+16). Per step each of the 8
// waves does one (gate, K-half) WMMA partial; LDS reduce; 16 threads run the gate math;
// monotonic-counter grid barrier publishes h[t+1].
__global__ void lstm_seq_kernel(const float* __restrict__ Wh,
                                const float* __restrict__ Zx,
                                const float* __restrict__ w_ph,
                                const float* __restrict__ b_p,
                                float* __restrict__ hbuf,      // (T+1) x H, hbuf[0]=0
                                float* __restrict__ out,
                                unsigned* __restrict__ counter,
                                unsigned* __restrict__ release)
{
    __shared__ float part[8][16];
    const int j0  = blockIdx.x << 4;
    const int w   = threadIdx.x >> 5;
    const int g   = w & 3;
    const int kh  = w >> 2;
    const int tid = threadIdx.x;

    float c_reg = 0.0f;   // c[j0+tid] for tid<16

    for (int t = 0; t < T_STEPS; ++t) {
        const float* hprev = hbuf + (size_t)t * HDIM;
        const float* Wrows = Wh + (((size_t)t * 4 + g) * HDIM + j0) * HDIM;

        v8f acc = wmma_matvec(Wrows, hprev, kh * 256, 256);
        __syncthreads();               // prior step's LDS reads complete
        store_col0(part[w], acc);
        __syncthreads();

        if (tid < 16) {
            const int m = tid;
            const size_t zo = (size_t)t * 4 * HDIM + j0 + m;
            float z0 = part[0][m] + part[4][m] + Zx[zo + 0 * HDIM];
            float z1 = part[1][m] + part[5][m] + Zx[zo + 1 * HDIM];
            float z2 = part[2][m] + part[6][m] + Zx[zo + 2 * HDIM];
            float z3 = part[3][m] + part[7][m] + Zx[zo + 3 * HDIM];
            float gt = tanhf(z0);
            float it = 1.0f / (1.0f + __expf(-z1));
            float ft = 1.0f / (1.0f + __expf(-z2));
            float ot = 1.0f / (1.0f + __expf(-z3));
            c_reg = gt * it + c_reg * ft;
            hbuf[(size_t)(t + 1) * HDIM + j0 + m] = tanhf(c_reg) * ot;
        }

        // ---- grid barrier (monotonic counter, sense via release generation) ----
        __syncthreads();
        if (tid == 0) {
            __threadfence();                               // publish h stores
            const unsigned target = (unsigned)(t + 1);
            unsigned prev = atomicAdd(counter, 1u);
            if (prev + 1u == (unsigned)NB_SEQ * target) {
                __hip_atomic_store(release, target, __ATOMIC_RELEASE,
                                   __HIP_MEMORY_SCOPE_AGENT);
            } else {
                while (__hip_atomic_load(release, __ATOMIC_ACQUIRE,
                                         __HIP_MEMORY_SCOPE_AGENT) < target)
                    __builtin_amdgcn_s_sleep(2);
            }
        }
        __syncthreads();
        __threadfence();                                   // acquire other WGPs' h
    }

    // ---- projection: out[j0+m] = w_ph[j0+m,:].h_T + b_p[j0+m] ----
    const float* hT    = hbuf + (size_t)T_STEPS * HDIM;
    const float* Wrows = w_ph + (size_t)j0 * HDIM;
    v8f acc = wmma_matvec(Wrows, hT, w * 64, 64);          // 8 waves cover K=512
    __syncthreads();
    store_col0(part[w], acc);
    __syncthreads();
    if (tid < 16) {
        float s = 0.0f;
        #pragma unroll
        for (int i = 0; i < 8; ++i) s += part[i][tid];
        out[j0 + tid] = s + b_p[j0 + tid];
    }
}

extern "C" void kernel_launch(void* const* d_in, const int* in_sizes, int n_in,
                              void* d_out, int out_size, void* d_ws, size_t ws_size,
                              hipStream_t stream)
{
    const float* x    = (const float*)d_in[0];   // [T,H]
    const float* Wx   = (const float*)d_in[1];   // [T,4,H,H]
    const float* Wh   = (const float*)d_in[2];   // [T,4,H,H]
    const float* b    = (const float*)d_in[3];   // [T,4,H]
    const float* w_ph = (const float*)d_in[4];   // [H,H]
    const float* b_p  = (const float*)d_in[5];   // [H]
    float* out = (float*)d_out;

    char* ws = (char*)d_ws;
    unsigned* counter = (unsigned*)ws;                 // barrier arrivals
    unsigned* release = (unsigned*)(ws + 64);          // barrier generation
    float* Zx   = (float*)(ws + 256);                                  // T*4*H f32 (1 MiB)
    float* hbuf = (float*)(ws + 256 + (size_t)T_STEPS * 4 * HDIM * 4); // (T+1)*H f32

    hipMemsetAsync(ws, 0, 256, stream);                      // zero barrier state
    hipMemsetAsync(hbuf, 0, HDIM * sizeof(float), stream);   // h_0 = 0

    zx_kernel<<<dim3(T_STEPS * 32), dim3(256), 0, stream>>>(x, Wx, b, Zx);
    lstm_seq_kernel<<<dim3(NB_SEQ), dim3(256), 0, stream>>>(Wh, Zx, w_ph, b_p,
                                                            hbuf, out, counter, release);
}